// LightweightTransformerLayer_47966194762204
// MI455X (gfx1250) — compile-verified
//
#include <hip/hip_runtime.h>
#include <math.h>

// ---------------- problem constants (match reference setup_inputs) ----------
#define DIMC     128
#define HEADS    4
#define HD       32          // head dim
#define WIN_HALF 32          // attend |i-j| <= 32
#define HIDDENC  256
#define LSEQ     2048
#define BATCH    4
#define NROWS    (BATCH * LSEQ)   // 8192
#define EPSLN    1e-5f
#define NEG_BIG  (-3.0e38f)

typedef __attribute__((ext_vector_type(2))) float v2f;
typedef __attribute__((ext_vector_type(8))) float v8f;

// D = A(16x4,f32) * B(4x16,f32) + C(16x16,f32), full fp32 precision.
static __device__ inline v8f wmma_f32(v2f a, v2f b, v8f c) {
  return __builtin_amdgcn_wmma_f32_16x16x4_f32(
      /*neg_a=*/false, a, /*neg_b=*/false, b,
      /*c_mod=*/(short)0, c, /*reuse_a=*/false, /*reuse_b=*/false);
}

// ---------------- layernorm: one wave (32 lanes) per 128-wide row -----------
__global__ __launch_bounds__(256) void k_layernorm(
    const float* __restrict__ x, const float* __restrict__ g,
    const float* __restrict__ b, float* __restrict__ out) {
  const int wave = threadIdx.x >> 5;
  const int lane = threadIdx.x & 31;
  const int row  = blockIdx.x * 8 + wave;
  const int c    = lane * 4;
  const float4 v = *(const float4*)(x + (size_t)row * DIMC + c);
  float s  = v.x + v.y + v.z + v.w;
  float s2 = v.x * v.x + v.y * v.y + v.z * v.z + v.w * v.w;
  #pragma unroll
  for (int m = 16; m >= 1; m >>= 1) {
    s  += __shfl_xor(s,  m, 32);
    s2 += __shfl_xor(s2, m, 32);
  }
  const float mu  = s * (1.0f / DIMC);
  const float var = s2 * (1.0f / DIMC) - mu * mu;
  const float inv = rsqrtf(var + EPSLN);
  const float4 gg = *(const float4*)(g + c);
  const float4 bb = *(const float4*)(b + c);
  float4 o;
  o.x = (v.x - mu) * inv * gg.x + bb.x;
  o.y = (v.y - mu) * inv * gg.y + bb.y;
  o.z = (v.z - mu) * inv * gg.z + bb.z;
  o.w = (v.w - mu) * inv * gg.w + bb.w;
  *(float4*)(out + (size_t)row * DIMC + c) = o;
}

// ---------------- fp32-WMMA GEMM, 16(M) x 64(N) tile per wave ---------------
// C[M,N] = A[M,K] @ W[K,N] (+bias)(gelu)(+res); lda=K, ldw=ldc=N.
// A tile is DMA'd to LDS with gfx1250 async-to-LDS loads (ASYNCcnt).
template <int K, bool HAS_BIAS, bool DO_GELU, bool HAS_RES>
__global__ __launch_bounds__(32) void k_gemm(
    const float* __restrict__ A, const float* __restrict__ W,
    const float* __restrict__ bias, const float* __restrict__ res,
    float* __restrict__ C, int N) {
  constexpr int LDA = K + 4;               // pad: frag reads hit 32 distinct banks
  __shared__ float As[16 * LDA];
  const int m0   = blockIdx.x * 16;
  const int n0   = blockIdx.y * 64;
  const int lane = threadIdx.x;
  const int half = lane >> 4;              // 0: K{0,1}/M 0..7 ; 1: K{2,3}/M 8..15
  const int lid  = lane & 15;
  const int koff = half * 2;

  // ---- async DMA the 16xK A tile into LDS (16B per lane per issue) ---------
  {
    constexpr int C4 = K / 4;              // float4 chunks per row
    #pragma unroll
    for (int c = lane; c < 16 * C4; c += 32) {
      const int row  = c / C4;
      const int col4 = c % C4;
      const unsigned lds = (unsigned)(uintptr_t)&As[row * LDA + col4 * 4];
      const unsigned long long g =
          (unsigned long long)(uintptr_t)(A + (size_t)(m0 + row) * K + col4 * 4);
      asm volatile("global_load_async_to_lds_b128 %0, %1, off"
                   :: "v"(lds), "v"(g) : "memory");
    }
    asm volatile("s_wait_asynccnt 0x0" ::: "memory");
    __syncthreads();
  }

  v8f acc[4];
  #pragma unroll
  for (int s = 0; s < 4; ++s) acc[s] = (v8f){0.f,0.f,0.f,0.f,0.f,0.f,0.f,0.f};

  const float* __restrict__ arow = &As[lid * LDA];
  for (int k = 0; k < K; k += 4) {
    v2f a; a.x = arow[k + koff];
           a.y = arow[k + koff + 1];
    const float* w0 = W + (size_t)(k + koff)     * N + n0 + lid;
    const float* w1 = W + (size_t)(k + koff + 1) * N + n0 + lid;
    #pragma unroll
    for (int s = 0; s < 4; ++s) {
      v2f b; b.x = w0[16 * s];
             b.y = w1[16 * s];
      acc[s] = wmma_f32(a, b, acc[s]);
    }
  }

  #pragma unroll
  for (int s = 0; s < 4; ++s) {
    const int col = n0 + 16 * s + lid;
    const float bcol = HAS_BIAS ? bias[col] : 0.0f;
    #pragma unroll
    for (int v = 0; v < 8; ++v) {
      const int row = m0 + v + 8 * half;
      float val = acc[s][v] + bcol;
      if (DO_GELU) val = 0.5f * val * (1.0f + erff(val * 0.70710678118654752f));
      if (HAS_RES) val += res[(size_t)row * N + col];
      C[(size_t)row * N + col] = val;
    }
  }
}

// ---------------- local windowed attention, one wave per (b,h,16 queries) ---
__global__ __launch_bounds__(32) void k_attn(const float* __restrict__ qkv,
                                             float* __restrict__ attn_out) {
  __shared__ float S[16 * 81];                 // P probabilities, padded stride
  const int idx  = blockIdx.x;
  const int qt   = idx & (LSEQ / 16 - 1);      // 0..127
  const int h    = (idx >> 7) & (HEADS - 1);
  const int b    = idx >> 9;
  const int i0   = qt * 16;
  const int jt0  = i0 - WIN_HALF;              // first key of the 80-key span
  const int lane = threadIdx.x;
  const int half = lane >> 4;
  const int lid  = lane & 15;
  const int koff = half * 2;
  const size_t baseRow = (size_t)b * LSEQ;
  const int ldq = 3 * DIMC;                    // 384

  // Q fragments for all 8 K-steps over head dim (kept in registers).
  v2f qa[8];
  {
    const float* qrow = qkv + (baseRow + i0 + lid) * (size_t)ldq + h * HD;
    #pragma unroll
    for (int t = 0; t < 8; ++t) {
      qa[t].x = qrow[4 * t + koff];
      qa[t].y = qrow[4 * t + koff + 1];
    }
  }

  // ---- scores: 5 key tiles of 16, masked to the +-32 window ----------------
  // Window validity is position-independent: with delta = t*16+lid-8*half-v,
  // valid <=> 0 <= delta <= 2*WIN_HALF, plus 0 <= j < L bounds.
  const float scale = 0.17677669529663688f;    // 1/sqrt(32)
  float sc[5][8];
  #pragma unroll
  for (int t = 0; t < 5; ++t) {
    const int j    = jt0 + t * 16 + lid;       // key col held by this lane
    const bool jin = (unsigned)j < (unsigned)LSEQ;
    const int jc   = j < 0 ? 0 : (j >= LSEQ ? LSEQ - 1 : j);
    const int dbase = t * 16 + lid - 8 * half;
    const float* krow = qkv + (baseRow + jc) * (size_t)ldq + DIMC + h * HD;
    v8f s = {0.f, 0.f, 0.f, 0.f, 0.f, 0.f, 0.f, 0.f};
    #pragma unroll
    for (int ks = 0; ks < 8; ++ks) {
      v2f bb; bb.x = krow[4 * ks + koff];
              bb.y = krow[4 * ks + koff + 1];
      s = wmma_f32(qa[ks], bb, s);
    }
    #pragma unroll
    for (int v = 0; v < 8; ++v) {
      const bool valid = jin && ((unsigned)(dbase - v) <= (unsigned)(2 * WIN_HALF));
      sc[t][v] = valid ? s[v] * scale : NEG_BIG;
    }
  }

  // ---- softmax per row: row lives in one 16-lane half (xor 1,2,4,8) --------
  #pragma unroll
  for (int v = 0; v < 8; ++v) {
    float mx = sc[0][v];
    #pragma unroll
    for (int t = 1; t < 5; ++t) mx = fmaxf(mx, sc[t][v]);
    #pragma unroll
    for (int m = 8; m >= 1; m >>= 1) mx = fmaxf(mx, __shfl_xor(mx, m, 32));
    float sm = 0.f;
    #pragma unroll
    for (int t = 0; t < 5; ++t) { sc[t][v] = __expf(sc[t][v] - mx); sm += sc[t][v]; }
    #pragma unroll
    for (int m = 8; m >= 1; m >>= 1) sm += __shfl_xor(sm, m, 32);
    const float inv = 1.0f / sm;
    #pragma unroll
    for (int t = 0; t < 5; ++t)
      S[(v + 8 * half) * 81 + t * 16 + lid] = sc[t][v] * inv;
  }
  __syncthreads();

  // ---- O = P(16x80) @ V(80x32), invalid keys contribute exactly zero -------
  v8f c0 = {0.f, 0.f, 0.f, 0.f, 0.f, 0.f, 0.f, 0.f};
  v8f c1 = {0.f, 0.f, 0.f, 0.f, 0.f, 0.f, 0.f, 0.f};
  for (int kk = 0; kk < 80; kk += 4) {
    v2f a; a.x = S[lid * 81 + kk + koff];
           a.y = S[lid * 81 + kk + koff + 1];
    int j0 = jt0 + kk + koff;
    int j1 = j0 + 1;
    const int jc0 = j0 < 0 ? 0 : (j0 >= LSEQ ? LSEQ - 1 : j0);
    const int jc1 = j1 < 0 ? 0 : (j1 >= LSEQ ? LSEQ - 1 : j1);
    const float* v0 = qkv + (baseRow + jc0) * (size_t)ldq + 2 * DIMC + h * HD;
    const float* v1 = qkv + (baseRow + jc1) * (size_t)ldq + 2 * DIMC + h * HD;
    v2f b0; b0.x = v0[lid];      b0.y = v1[lid];
    c0 = wmma_f32(a, b0, c0);
    v2f b1; b1.x = v0[16 + lid]; b1.y = v1[16 + lid];
    c1 = wmma_f32(a, b1, c1);
  }
  #pragma unroll
  for (int v = 0; v < 8; ++v) {
    const size_t row = baseRow + i0 + v + 8 * half;
    attn_out[row * DIMC + h * HD + lid]      = c0[v];
    attn_out[row * DIMC + h * HD + 16 + lid] = c1[v];
  }
}

// ---------------- launcher ---------------------------------------------------
extern "C" void kernel_launch(void* const* d_in, const int* in_sizes, int n_in,
                              void* d_out, int out_size, void* d_ws, size_t ws_size,
                              hipStream_t stream) {
  const float* x      = (const float*)d_in[0];
  const float* w_qkv  = (const float*)d_in[1];
  const float* w_proj = (const float*)d_in[2];
  const float* ln1_g  = (const float*)d_in[3];
  const float* ln1_b  = (const float*)d_in[4];
  const float* ln2_g  = (const float*)d_in[5];
  const float* ln2_b  = (const float*)d_in[6];
  const float* w1     = (const float*)d_in[7];
  const float* b1     = (const float*)d_in[8];
  const float* w2     = (const float*)d_in[9];
  const float* b2     = (const float*)d_in[10];
  float* out = (float*)d_out;

  float* ws    = (float*)d_ws;
  float* lnbuf = ws;                                   // 8192*128
  float* qkv   = lnbuf + (size_t)NROWS * DIMC;         // 8192*384
  float* attn  = qkv + (size_t)NROWS * 3 * DIMC;       // 8192*128
  float* x1    = attn + (size_t)NROWS * DIMC;          // 8192*128
  float* hmid  = qkv;    // qkv dead after attn; reuse (8192*256 <= 8192*384)

  // 1. ln1 = LN(x)
  k_layernorm<<<NROWS / 8, 256, 0, stream>>>(x, ln1_g, ln1_b, lnbuf);
  // 2. qkv = ln1 @ w_qkv            (N=384 -> 6 column tiles of 64)
  k_gemm<DIMC, false, false, false><<<dim3(NROWS / 16, 6), 32, 0, stream>>>(
      lnbuf, w_qkv, nullptr, nullptr, qkv, 3 * DIMC);
  // 3. windowed attention
  k_attn<<<BATCH * HEADS * (LSEQ / 16), 32, 0, stream>>>(qkv, attn);
  // 4. x1 = x + attn @ w_proj       (N=128 -> 2 column tiles)
  k_gemm<DIMC, false, false, true><<<dim3(NROWS / 16, 2), 32, 0, stream>>>(
      attn, w_proj, nullptr, x, x1, DIMC);
  // 5. ln2 = LN(x1)
  k_layernorm<<<NROWS / 8, 256, 0, stream>>>(x1, ln2_g, ln2_b, lnbuf);
  // 6. hmid = gelu(ln2 @ w1 + b1)   (N=256 -> 4 column tiles)
  k_gemm<DIMC, true, true, false><<<dim3(NROWS / 16, 4), 32, 0, stream>>>(
      lnbuf, w1, b1, nullptr, hmid, HIDDENC);
  // 7. out = x1 + hmid @ w2 + b2    (K=256, N=128 -> 2 column tiles)
  k_gemm<HIDDENC, true, false, true><<<dim3(NROWS / 16, 2), 32, 0, stream>>>(
      hmid, w2, b2, x1, out, DIMC);
}